// TieredMemoryModule_35270271435143
// MI455X (gfx1250) — compile-verified
//
#include <hip/hip_runtime.h>
#include <hip/hip_bf16.h>

#define HID   256
#define FASTN 64
#define SLOWN 256
#define VOCAB 64
#define BATCH 64
#define TLEN  2048
#define NEGC  (-1000000000.0f)

typedef __attribute__((ext_vector_type(2))) float v2f;
typedef __attribute__((ext_vector_type(8))) float v8f;

// ---------------------------------------------------------------------------
// Kernel 1: per-vocab lookup tables: gate[v]=sigmoid(emb[v].Wg+bg),
//           demsc[v]=emb[v].Wd+bd.  One wave per vocab entry.
// ---------------------------------------------------------------------------
__global__ __launch_bounds__(32)
void tm_tables_kernel(const float* __restrict__ emb,
                      const float* __restrict__ Wg, const float* __restrict__ bg,
                      const float* __restrict__ Wd, const float* __restrict__ bd,
                      float* __restrict__ gate, float* __restrict__ demsc) {
    const int v = blockIdx.x;
    const int lane = threadIdx.x;
    float sg = 0.0f, sd = 0.0f;
    for (int k = lane; k < HID; k += 32) {
        const float e = emb[v * HID + k];
        sg += e * Wg[k];
        sd += e * Wd[k];
    }
    #pragma unroll
    for (int off = 16; off > 0; off >>= 1) {
        sg += __shfl_xor(sg, off, 32);
        sd += __shfl_xor(sd, off, 32);
    }
    if (lane == 0) {
        const float x = sg + bg[0];
        gate[v]  = 1.0f / (1.0f + expf(-x));
        demsc[v] = sd + bd[0];
    }
}

// ---------------------------------------------------------------------------
// Kernel 2: fp32 WMMA GEMM, fully specialized at compile time.
//   C[M,N] = A[M,K] @ B + bias   (B is [K,N], or [N,K] if TRANSB)
// One wave computes one 16x16 C tile with V_WMMA_F32_16X16X4_F32.
// K-loop processes 16 K-values per trip: 8 fragment loads issued up front
// (pairs merge to global_load_b64 / form a clause), then 4 wmmas, so waits
// drain progressively instead of load->wait0->wmma serialization.
// Fragment layouts (per CDNA5 ISA):
//   A: lane holds A[m=lane&15][kk + r + 2*(lane>>4)] in vgpr r (r=0,1)
//   B: lane holds B[kk + r + 2*(lane>>4)][tn + (lane&15)]
//   C: vgpr i holds C[tm + i + 8*(lane>>4)][tn + (lane&15)]
// ---------------------------------------------------------------------------
template<int N, int K, bool TRANSB, bool HASBIAS>
__global__ __launch_bounds__(32)
void wmma_gemm_f32_t(const float* __restrict__ A, const float* __restrict__ B,
                     const float* __restrict__ bias, float* __restrict__ C) {
    const int lane = threadIdx.x;
    const int m16  = lane & 15;
    const int g    = lane >> 4;                 // 0 or 1
    const int col  = blockIdx.x * 16 + m16;
    const int tm   = blockIdx.y * 16;
    const int rowA = tm + m16;

    v8f acc;
    const float bv = HASBIAS ? bias[col] : 0.0f;
    #pragma unroll
    for (int i = 0; i < 8; ++i) acc[i] = bv;

    const float* ap = A + (size_t)rowA * K + 2 * g;
    const float* bp = TRANSB ? (B + (size_t)col * K + 2 * g)
                             : (B + (size_t)(2 * g) * N + col);

    static_assert(K % 16 == 0, "K must be a multiple of 16");
    for (int kk = 0; kk < K; kk += 16) {
        v2f a[4], b[4];
        #pragma unroll
        for (int j = 0; j < 4; ++j) {
            a[j].x = ap[4 * j + 0];
            a[j].y = ap[4 * j + 1];
            if (TRANSB) {
                b[j].x = bp[4 * j + 0];
                b[j].y = bp[4 * j + 1];
            } else {
                b[j].x = bp[(size_t)(4 * j + 0) * N];
                b[j].y = bp[(size_t)(4 * j + 1) * N];
            }
        }
        #pragma unroll
        for (int j = 0; j < 4; ++j)
            acc = __builtin_amdgcn_wmma_f32_16x16x4_f32(
                /*neg_a=*/false, a[j], /*neg_b=*/false, b[j],
                /*c_mod=*/(short)0, acc, /*reuse_a=*/false, /*reuse_b=*/false);
        ap += 16;
        bp += TRANSB ? (size_t)16 : (size_t)16 * N;
    }
    #pragma unroll
    for (int i = 0; i < 8; ++i)
        C[(size_t)(tm + i + 8 * g) * N + col] = acc[i];
}

// ---------------------------------------------------------------------------
// Kernel 3: sequential tiered-memory scan, one wave32 per batch row.
// VOCAB=64 collapses all per-step linear algebra to 64-entry lookup tables;
// state (token ids, demotion scores, LRU timestamps) lives in LDS; eviction
// argmin/argmax are wave shuffle reductions with first-index tie-break.
// Afterwards: masked softmax over the 320 memory slots, attention mass
// binned per vocab id into wbin[b][64]; also writes slow_mask.
// ---------------------------------------------------------------------------
__global__ __launch_bounds__(32)
void tm_scan_kernel(const int* __restrict__ seq,
                    const float* __restrict__ gate, const float* __restrict__ demsc,
                    const float* __restrict__ SV,            // [VOCAB mem][VOCAB last]
                    float* __restrict__ slow_mask_out,       // [B][SLOWN]
                    float* __restrict__ wbin) {              // [B][VOCAB]
    __shared__ float sGate[VOCAB];
    __shared__ float sDem[VOCAB];
    __shared__ int   sFastTok[FASTN];
    __shared__ float sFastDem[FASTN];
    __shared__ int   sSlowTok[SLOWN];
    __shared__ int   sSlowTs[SLOWN];
    __shared__ float sW[VOCAB];

    const int b    = blockIdx.x;
    const int lane = threadIdx.x;

    sGate[lane]      = gate[lane];
    sGate[lane + 32] = gate[lane + 32];
    sDem[lane]       = demsc[lane];
    sDem[lane + 32]  = demsc[lane + 32];
    __syncthreads();

    const int* srow = seq + b * TLEN;
    int fastCount = 0, slowCount = 0;   // uniform across the wave

    for (int t = 0; t < TLEN - 3; ++t) {
        const int tok = srow[t];                 // wave-uniform
        const float w = sGate[tok];
        if (w >= 0.4f) {                         // uniform branch
            int fslot;
            int dtok = 0;
            bool doSlow = false;
            if (fastCount < FASTN) {
                fslot = fastCount++;
            } else {
                // argmin over 64 demotion scores, first-index tie-break
                float bs  = sFastDem[lane];
                int   bid = lane;
                const float s1 = sFastDem[lane + 32];
                if (s1 < bs) { bs = s1; bid = lane + 32; }
                #pragma unroll
                for (int off = 16; off > 0; off >>= 1) {
                    const float os  = __shfl_xor(bs, off, 32);
                    const int   oid = __shfl_xor(bid, off, 32);
                    if (os < bs || (os == bs && oid < bid)) { bs = os; bid = oid; }
                }
                fslot  = bid;
                dtok   = sFastTok[fslot];
                doSlow = true;
            }
            if (doSlow) {
                int ss;
                if (slowCount < SLOWN) {
                    ss = slowCount++;
                } else {
                    // LRU: argmin write-timestamp over 256 slots
                    int bts = sSlowTs[lane];
                    int bid = lane;
                    #pragma unroll
                    for (int k = 1; k < 8; ++k) {
                        const int sidx = lane + 32 * k;
                        const int tsv  = sSlowTs[sidx];
                        if (tsv < bts) { bts = tsv; bid = sidx; }
                    }
                    #pragma unroll
                    for (int off = 16; off > 0; off >>= 1) {
                        const int ots = __shfl_xor(bts, off, 32);
                        const int oid = __shfl_xor(bid, off, 32);
                        if (ots < bts || (ots == bts && oid < bid)) { bts = ots; bid = oid; }
                    }
                    ss = bid;
                }
                if (lane == 0) { sSlowTok[ss] = dtok; sSlowTs[ss] = t; }
            }
            if (lane == 0) { sFastTok[fslot] = tok; sFastDem[fslot] = sDem[tok]; }
            __syncthreads();
        }
    }
    __syncthreads();

    // ------- read head: gather per-slot scores, masked softmax, vocab binning
    const int lt = srow[TLEN - 1];
    float sc[10];
    int   tk[10];
    bool  us[10];
    #pragma unroll
    for (int i = 0; i < 2; ++i) {               // fast slots lane, lane+32
        const int n = lane + 32 * i;
        const bool u = (n < fastCount);
        int tokv = 0; float s = NEGC;
        if (u) { tokv = sFastTok[n]; s = SV[tokv * VOCAB + lt]; }
        sc[i] = s; tk[i] = tokv; us[i] = u;
    }
    #pragma unroll
    for (int k = 0; k < 8; ++k) {               // slow slots lane+32k
        const int m = lane + 32 * k;
        const bool u = (m < slowCount);
        int tokv = 0; float s = NEGC;
        if (u) { tokv = sSlowTok[m]; s = SV[tokv * VOCAB + lt]; }
        sc[2 + k] = s; tk[2 + k] = tokv; us[2 + k] = u;
    }
    float mx = NEGC;
    #pragma unroll
    for (int i = 0; i < 10; ++i) mx = fmaxf(mx, sc[i]);
    #pragma unroll
    for (int off = 16; off > 0; off >>= 1) mx = fmaxf(mx, __shfl_xor(mx, off, 32));

    float e[10];
    float se = 0.0f;
    #pragma unroll
    for (int i = 0; i < 10; ++i) { e[i] = expf(sc[i] - mx); se += e[i]; }
    #pragma unroll
    for (int off = 16; off > 0; off >>= 1) se += __shfl_xor(se, off, 32);
    const float inv = 1.0f / se;

    sW[lane] = 0.0f; sW[lane + 32] = 0.0f;
    __syncthreads();
    #pragma unroll
    for (int i = 0; i < 10; ++i)
        if (us[i]) atomicAdd(&sW[tk[i]], e[i] * inv);
    __syncthreads();

    wbin[b * VOCAB + lane]      = sW[lane];
    wbin[b * VOCAB + lane + 32] = sW[lane + 32];
    #pragma unroll
    for (int k = 0; k < 8; ++k) {
        const int m = lane + 32 * k;
        slow_mask_out[b * SLOWN + m] = (m < slowCount) ? 1.0f : 0.0f;
    }
}

// ---------------------------------------------------------------------------
extern "C" void kernel_launch(void* const* d_in, const int* in_sizes, int n_in,
                              void* d_out, int out_size, void* d_ws, size_t ws_size,
                              hipStream_t stream) {
    (void)in_sizes; (void)n_in; (void)out_size; (void)ws_size;
    const int*   seq = (const int*)d_in[0];
    const float* emb = (const float*)d_in[1];
    const float* Wg  = (const float*)d_in[2];
    const float* bg  = (const float*)d_in[3];
    const float* Wd  = (const float*)d_in[4];
    const float* bd  = (const float*)d_in[5];
    const float* Wq  = (const float*)d_in[6];
    const float* bq  = (const float*)d_in[7];
    const float* Wo  = (const float*)d_in[8];
    const float* bo  = (const float*)d_in[9];

    float* out = (float*)d_out;                 // [0,4096) logits, [4096,20480) slow_mask
    float* ws  = (float*)d_ws;
    float* gate  = ws;                          // 64
    float* demsc = ws + 64;                     // 64
    float* qtab  = ws + 128;                    // 64*256
    float* SV    = qtab + VOCAB * HID;          // 64*64
    float* wbin  = SV + VOCAB * VOCAB;          // 64*64
    float* ctx   = wbin + BATCH * VOCAB;        // 64*256

    // 1) lookup tables
    tm_tables_kernel<<<VOCAB, 32, 0, stream>>>(emb, Wg, bg, Wd, bd, gate, demsc);
    // 2) qtab = emb @ Wq + bq            (64x256 = 64x256 @ 256x256)
    wmma_gemm_f32_t<HID, HID, false, true>
        <<<dim3(HID / 16, VOCAB / 16), 32, 0, stream>>>(emb, Wq, bq, qtab);
    // 3) SV = emb @ qtab^T               (64x64  = 64x256 @ (64x256)^T)
    wmma_gemm_f32_t<VOCAB, HID, true, false>
        <<<dim3(VOCAB / 16, VOCAB / 16), 32, 0, stream>>>(emb, qtab, nullptr, SV);
    // 4) sequential scan + softmax + vocab binning (+ slow_mask output)
    tm_scan_kernel<<<BATCH, 32, 0, stream>>>(seq, gate, demsc, SV,
                                             out + BATCH * VOCAB, wbin);
    // 5) ctx = wbin @ emb                (64x256 = 64x64 @ 64x256)
    wmma_gemm_f32_t<HID, VOCAB, false, false>
        <<<dim3(HID / 16, BATCH / 16), 32, 0, stream>>>(wbin, emb, nullptr, ctx);
    // 6) logits = ctx @ Wo + bo          (64x64  = 64x256 @ 256x64)
    wmma_gemm_f32_t<VOCAB, HID, false, true>
        <<<dim3(VOCAB / 16, BATCH / 16), 32, 0, stream>>>(ctx, Wo, bo, out);
}